// filter_layer_33801392620108
// MI455X (gfx1250) — compile-verified
//
#include <hip/hip_runtime.h>
#include <cstdint>

typedef float v2f __attribute__((ext_vector_type(2)));
typedef float v8f __attribute__((ext_vector_type(8)));

#define BATCH   64
#define NEURON  1024
#define STEPS   2048
#define CHUNK   16
#define NCHUNK  (STEPS / CHUNK)
#define WAVES_PER_BLOCK 8
#define TILE_SEQ 16
#define LDS_STRIDE 20   // floats per sequence row: 16 data + 4 pad (keeps 16B row alignment)

// y_t = v0*(m_t - s_t);  m_t = dm*m_{t-1}+x_t;  s_t = ds*s_{t-1}+x_t
// Chunked as Y = H @ X + carry, H[j][i] = v0*(dm^{j-i} - ds^{j-i}) lower-triangular Toeplitz.
__global__ __launch_bounds__(256) void snn_filter_wmma(const float* __restrict__ in,
                                                       float* __restrict__ out) {
    __shared__ float smem[WAVES_PER_BLOCK][2][TILE_SEQ * LDS_STRIDE];

    const int lane = threadIdx.x & 31;
    const int wave = threadIdx.x >> 5;
    const int h    = lane >> 4;        // lane half (selects K pairing in A/B layouts)
    const int m    = lane & 15;        // A row index / B,C,D column index

    const float v0 = 2.1165347359575995f;   // eta^(eta/(eta-1))/(eta-1), eta = 4

    auto powdm = [](int d) { return __expf((float)d * -0.05f); }; // dm^d, dm = e^{-1/20}
    auto powds = [](int d) { return __expf((float)d * -0.2f);  }; // ds^d, ds = e^{-1/5}

    // A operand: four 16x4 slabs of H (fp32, 2 VGPRs each; VGPR0 = K0|K2, VGPR1 = K1|K3)
    v2f A[4];
#pragma unroll
    for (int kk = 0; kk < 4; ++kk) {
        int k0 = 4 * kk + 2 * h;
        int d0 = m - k0, d1 = m - (k0 + 1);
        A[kk].x = (d0 >= 0) ? v0 * (powdm(d0) - powds(d0)) : 0.0f;
        A[kk].y = (d1 >= 0) ? v0 * (powdm(d1) - powds(d1)) : 0.0f;
    }
    // carry-in factors per accumulator row j = r + 8h:  v0*dm^{j+1}, v0*ds^{j+1}
    float fm[8], fs[8];
#pragma unroll
    for (int r = 0; r < 8; ++r) {
        int j = r + 8 * h;
        fm[r] = v0 * powdm(j + 1);
        fs[r] = v0 * powds(j + 1);
    }
    // carry-update weights for this lane's held B elements (K = 4kk + 2h + e)
    float wmv[8], wsv[8];
#pragma unroll
    for (int kk = 0; kk < 4; ++kk)
#pragma unroll
        for (int e = 0; e < 2; ++e) {
            int k = 4 * kk + 2 * h + e;
            wmv[kk * 2 + e] = powdm(15 - k);
            wsv[kk * 2 + e] = powds(15 - k);
        }
    const float dm16 = powdm(16), ds16 = powds(16);

    // this wave's 16 sequences
    const int tile = blockIdx.x * WAVES_PER_BLOCK + wave;
    const size_t base = (size_t)tile * TILE_SEQ * STEPS;
    const uint64_t gin = (uint64_t)(uintptr_t)(in + base);
    float* outp_base = out + base;

    const unsigned ldsbuf0 = (unsigned)(uintptr_t)(&smem[wave][0][0]); // LDS byte offset
    const unsigned ldsbuf1 = ldsbuf0 + TILE_SEQ * LDS_STRIDE * 4;

    const int n2 = lane >> 2;        // 0..7  (sequence for coalesced 16B transfers)
    const int t4 = (lane & 3) * 4;   // 0,4,8,12 (time offset)

    // global -> LDS async DMA of one 16x16 chunk, [seq][t] layout in LDS
#define ISSUE_LOAD(cc, ldsb)                                                     \
    do {                                                                         \
        unsigned l0 = (ldsb) + (unsigned)(n2 * LDS_STRIDE * 4 + t4 * 4);         \
        unsigned l1 = l0 + 8u * LDS_STRIDE * 4;                                  \
        unsigned g0 = (unsigned)(n2 * (STEPS * 4) + (cc) * (CHUNK * 4) + t4 * 4);\
        unsigned g1 = g0 + 8u * (STEPS * 4);                                     \
        asm volatile("global_load_async_to_lds_b128 %0, %1, %2"                  \
                     :: "v"(l0), "v"(g0), "s"(gin) : "memory");                  \
        asm volatile("global_load_async_to_lds_b128 %0, %1, %2"                  \
                     :: "v"(l1), "v"(g1), "s"(gin) : "memory");                  \
    } while (0)

    ISSUE_LOAD(0, ldsbuf0);

    float Mc = 0.0f, Sc = 0.0f;
    unsigned curbuf = ldsbuf0, nxtbuf = ldsbuf1;
    int p = 0;

    for (int c = 0; c < NCHUNK; ++c) {
        if (c + 1 < NCHUNK) {
            // prior iteration's DS reads of nxtbuf must retire before DMA overwrites it
            asm volatile("s_wait_dscnt 0x0" ::: "memory");
            ISSUE_LOAD(c + 1, nxtbuf);
            asm volatile("s_wait_asynccnt 0x1" ::: "memory"); // chunk c arrived
        } else {
            asm volatile("s_wait_asynccnt 0x0" ::: "memory");
        }

        const float* Xb = &smem[wave][p][0];
        // B operand: ds_load_b64 per K-quad; VGPR0 = K0|K2, VGPR1 = K1|K3, N = lane&15
        v2f b0 = *(const v2f*)(Xb + m * LDS_STRIDE + 0  + 2 * h);
        v2f b1 = *(const v2f*)(Xb + m * LDS_STRIDE + 4  + 2 * h);
        v2f b2 = *(const v2f*)(Xb + m * LDS_STRIDE + 8  + 2 * h);
        v2f b3 = *(const v2f*)(Xb + m * LDS_STRIDE + 12 + 2 * h);

        // seed accumulator with decayed carry contribution
        v8f acc;
#pragma unroll
        for (int r = 0; r < 8; ++r) acc[r] = fm[r] * Mc - fs[r] * Sc;

        acc = __builtin_amdgcn_wmma_f32_16x16x4_f32(false, A[0], false, b0, (short)0, acc, false, false);
        acc = __builtin_amdgcn_wmma_f32_16x16x4_f32(false, A[1], false, b1, (short)0, acc, false, false);
        acc = __builtin_amdgcn_wmma_f32_16x16x4_f32(false, A[2], false, b2, (short)0, acc, false, false);
        acc = __builtin_amdgcn_wmma_f32_16x16x4_f32(false, A[3], false, b3, (short)0, acc, false, false);

        // carry update: per-lane weighted partial sums, combine the two lane halves
        float pm = wmv[0]*b0.x + wmv[1]*b0.y + wmv[2]*b1.x + wmv[3]*b1.y
                 + wmv[4]*b2.x + wmv[5]*b2.y + wmv[6]*b3.x + wmv[7]*b3.y;
        float ps = wsv[0]*b0.x + wsv[1]*b0.y + wsv[2]*b1.x + wsv[3]*b1.y
                 + wsv[4]*b2.x + wsv[5]*b2.y + wsv[6]*b3.x + wsv[7]*b3.y;
        pm += __shfl_xor(pm, 16, 32);
        ps += __shfl_xor(ps, 16, 32);
        Mc = dm16 * Mc + pm;
        Sc = ds16 * Sc + ps;

        // transpose Y through the (now consumed) current buffer, store coalesced
        float* Yb = &smem[wave][p][0];
#pragma unroll
        for (int r = 0; r < 8; ++r) Yb[m * LDS_STRIDE + (r + 8 * h)] = acc[r];

        asm volatile("s_wait_dscnt 0x0" ::: "memory"); // cross-lane LDS visibility (wave-local)

        float4 y0 = *(const float4*)(Yb + n2 * LDS_STRIDE + t4);
        float4 y1 = *(const float4*)(Yb + (n2 + 8) * LDS_STRIDE + t4);
        float* op = outp_base + c * CHUNK;
        *(float4*)(op + n2 * STEPS + t4)       = y0;
        *(float4*)(op + (n2 + 8) * STEPS + t4) = y1;

        p ^= 1;
        unsigned tb = curbuf; curbuf = nxtbuf; nxtbuf = tb;
    }
#undef ISSUE_LOAD
}

extern "C" void kernel_launch(void* const* d_in, const int* in_sizes, int n_in,
                              void* d_out, int out_size, void* d_ws, size_t ws_size,
                              hipStream_t stream) {
    (void)in_sizes; (void)n_in; (void)out_size; (void)d_ws; (void)ws_size;
    const float* in = (const float*)d_in[0];
    float* out = (float*)d_out;
    dim3 grid((BATCH * NEURON) / (TILE_SEQ * WAVES_PER_BLOCK)); // 512 blocks
    snn_filter_wmma<<<grid, 256, 0, stream>>>(in, out);
}